// GAT_15994458210578
// MI455X (gfx1250) — compile-verified
//
#include <hip/hip_runtime.h>
#include <hip/hip_bf16.h>
#include <math.h>

// ---------------- problem constants (from reference) ----------------
#define NPG   62                      // nodes per graph
#define NG    4096                    // graphs
#define NN    (NG * NPG)              // 253952 nodes
#define DEG   16                      // in-edges per node (plus 1 self loop)
#define FI    32                      // input features
#define HID   64                      // hidden features
#define BN_EPS 1e-5f
#define SM_EPS 1e-16f

typedef __attribute__((ext_vector_type(2))) float v2f;
typedef __attribute__((ext_vector_type(8))) float v8f;

// =====================================================================
// Kernel 1: GATv2 layer 1 (one graph per block).
//   phase 1: xl = x@Wl1+bl, xr = x@Wr1+br      (fp32 WMMA, 64 ops/wave)
//   phase 2: edge logits alpha = att.lrelu(xl[s]+xr[i])
//   phase 3: per-dst softmax -> dense attention matrix P (62x62, padded)
//   phase 4: h = P @ xl                        (fp32 WMMA, 64 ops/wave)
//   phase 5: global write + per-graph BN partial sums
// =====================================================================
__global__ __launch_bounds__(128) void gat1_kernel(
    const float* __restrict__ x,      // [N, 32]
    const int*   __restrict__ esrc,   // edge_index[0] : [E] src node ids
    const float* __restrict__ Wl1,    // [32, 64]
    const float* __restrict__ bl1,    // [64]
    const float* __restrict__ Wr1,    // [32, 64]
    const float* __restrict__ br1,    // [64]
    const float* __restrict__ att1,   // [64]
    const float* __restrict__ bias1,  // [64]
    float* __restrict__ hout,         // [N, 64]  (workspace)
    float* __restrict__ psum,         // [NG, 64] partial sums
    float* __restrict__ psq)          // [NG, 64] partial sums of squares
{
    // pooled LDS, phases alias dead regions:
    //   phase 1 : xs [64][33] (2112 f) | W [2][32][64] (4096 f)
    //   phase 2+: al [62*17] (1054 f) | sl ushort[62*17] | P [64][65] @ +2048
    __shared__ float pool[6208];
    __shared__ float hsh[2][64][65];      // [0]=xl, [1]=xr (later: h)
    __shared__ float attS[64], bS0[64], bS1[64], bias1S[64];
    __shared__ float red[128];

    float*          xs = pool;                            // [64][33]
    float*          W  = pool + 2112;                     // [2][32][64]
    float*          al = pool;                            // [62*17]
    unsigned short* sl = (unsigned short*)(pool + 1054);  // [62*17]
    float*          P  = pool + 2048;                     // [64][65]

    const int g    = blockIdx.x;
    const int tid  = threadIdx.x;
    const int base = g * NPG;

    // ---- stage inputs ----
    for (int t = tid; t < 64 * FI; t += 128) {
        int r = t >> 5, c = t & 31;
        xs[r * 33 + c] = (r < NPG) ? x[(size_t)(base + r) * FI + c] : 0.0f;
    }
    for (int t = tid; t < FI * HID; t += 128) {
        W[t]        = Wl1[t];
        W[2048 + t] = Wr1[t];
    }
    if (tid < 64) {
        attS[tid]   = att1[tid];
        bS0[tid]    = bl1[tid];
        bS1[tid]    = br1[tid];
        bias1S[tid] = bias1[tid];
    }
    __syncthreads();

    // ---- phase 1: two 64x64x32 matmuls via V_WMMA_F32_16X16X4_F32 ----
    const int wave = tid >> 5;             // M-tile owner
    const int lane = tid & 31;
    const int lr   = lane & 15;
    const int lhi  = lane >> 4;            // K pair select
    const int arow = wave * 16 + lr;

#pragma unroll
    for (int m = 0; m < 2; ++m) {
#pragma unroll
        for (int n = 0; n < 4; ++n) {
            v8f acc = {};
#pragma unroll
            for (int k = 0; k < 8; ++k) {
                const int kk = k * 4 + lhi * 2;
                v2f a, b;
                a.x = xs[arow * 33 + kk];
                a.y = xs[arow * 33 + kk + 1];
                b.x = W[m * 2048 + kk * 64 + n * 16 + lr];
                b.y = W[m * 2048 + (kk + 1) * 64 + n * 16 + lr];
                acc = __builtin_amdgcn_wmma_f32_16x16x4_f32(
                        false, a, false, b, (short)0, acc, false, false);
            }
            const int   col  = n * 16 + lr;
            const float bias = (m == 0) ? bS0[col] : bS1[col];
#pragma unroll
            for (int r = 0; r < 8; ++r)
                hsh[m][wave * 16 + r + lhi * 8][col] = acc[r] + bias;
        }
    }
    __syncthreads();   // xs / W dead; al/sl/P may overwrite pool

    // ---- phase 2: edge attention logits; zero P ----
    for (int t = tid; t < 64 * 65; t += 128) P[t] = 0.0f;
    for (int t = tid; t < NPG * 17; t += 128) {
        const int i = t / 17, j = t - i * 17;
        const int s = (j < DEG) ? (esrc[(size_t)(base + i) * DEG + j] - base) : i;
        sl[t] = (unsigned short)s;
        float acc = 0.0f;
        for (int c = 0; c < HID; ++c) {
            float v = hsh[0][s][c] + hsh[1][i][c];
            v = (v > 0.0f) ? v : 0.2f * v;
            acc += attS[c] * v;
        }
        al[t] = acc;
    }
    __syncthreads();

    // ---- phase 3: per-dst softmax -> scatter normalized weights into P ----
    if (tid < NPG) {
        float mx = -3.4e38f;
        for (int j = 0; j < 17; ++j) mx = fmaxf(mx, al[tid * 17 + j]);
        float den = 0.0f;
        for (int j = 0; j < 17; ++j) {
            const float e = expf(al[tid * 17 + j] - mx);
            al[tid * 17 + j] = e;                 // row owned by this thread
            den += e;
        }
        const float inv = 1.0f / (den + SM_EPS);
        for (int j = 0; j < 17; ++j)              // repeated src accumulate
            P[tid * 65 + sl[tid * 17 + j]] += al[tid * 17 + j] * inv;
    }
    __syncthreads();

    // ---- phase 4: h = P @ xl via WMMA (M=dst, K=src, N=channel) ----
#pragma unroll
    for (int n = 0; n < 4; ++n) {
        v8f acc = {};
#pragma unroll
        for (int k = 0; k < 16; ++k) {
            const int kk = k * 4 + lhi * 2;
            v2f a, b;
            a.x = P[arow * 65 + kk];
            a.y = P[arow * 65 + kk + 1];
            b.x = hsh[0][kk][n * 16 + lr];
            b.y = hsh[0][kk + 1][n * 16 + lr];
            acc = __builtin_amdgcn_wmma_f32_16x16x4_f32(
                    false, a, false, b, (short)0, acc, false, false);
        }
        const int col = n * 16 + lr;
#pragma unroll
        for (int r = 0; r < 8; ++r)
            hsh[1][wave * 16 + r + lhi * 8][col] = acc[r] + bias1S[col];
    }
    __syncthreads();

    // ---- phase 5: coalesced global write + BN partials ----
    float lsum = 0.0f, lsq = 0.0f;
    const int c = tid & 63;
    for (int i = tid >> 6; i < NPG; i += 2) {
        const float v = hsh[1][i][c];
        hout[(size_t)(base + i) * HID + c] = v;
        lsum += v;
        lsq  += v * v;
    }
    red[tid] = lsum;
    __syncthreads();
    if (tid < 64) psum[(size_t)g * HID + tid] = red[tid] + red[tid + 64];
    __syncthreads();
    red[tid] = lsq;
    __syncthreads();
    if (tid < 64) psq[(size_t)g * HID + tid] = red[tid] + red[tid + 64];
}

// =====================================================================
// Kernel 2: finalize BatchNorm statistics -> per-channel scale/shift
// =====================================================================
__global__ __launch_bounds__(128) void bnstat_kernel(
    const float* __restrict__ psum, const float* __restrict__ psq,
    const float* __restrict__ gamma, const float* __restrict__ beta,
    float* __restrict__ bn)           // [128]: scale[64] | shift[64]
{
    const int t = threadIdx.x;        // 0..127
    const int c = t & 63;
    const float* src = (t < 64) ? psum : psq;
    float s = 0.0f;
    for (int g = 0; g < NG; ++g) s += src[(size_t)g * HID + c];
    __shared__ float sm[128];
    sm[t] = s;
    __syncthreads();
    if (t < 64) {
        const float mean = sm[t] * (1.0f / (float)NN);
        const float var  = sm[t + 64] * (1.0f / (float)NN) - mean * mean;
        const float inv  = rsqrtf(var + BN_EPS);
        const float sc   = gamma[t] * inv;
        bn[t]      = sc;
        bn[64 + t] = beta[t] - mean * sc;
    }
}

// =====================================================================
// Kernel 3: BN apply + GATv2 layer 2 (HID->1) + exact GELU
// =====================================================================
__global__ __launch_bounds__(128) void gat2_kernel(
    const float* __restrict__ hin,    // [N, 64]
    const int*   __restrict__ esrc,
    const float* __restrict__ bn,     // scale[64] | shift[64]
    const float* __restrict__ Wl2,    // [64]
    const float* __restrict__ bl2,    // [1]
    const float* __restrict__ Wr2,    // [64]
    const float* __restrict__ br2,    // [1]
    const float* __restrict__ att2,   // [1]
    const float* __restrict__ bias2,  // [1]
    float* __restrict__ out)          // [NG, 62]
{
    __shared__ float hn[NPG][65];
    __shared__ float yl[64], yr[64];
    __shared__ float al[NPG][17];
    __shared__ unsigned short sl[NPG][17];
    __shared__ float wl[64], wr[64];

    const int g    = blockIdx.x;
    const int tid  = threadIdx.x;
    const int base = g * NPG;

    if (tid < 64) { wl[tid] = Wl2[tid]; wr[tid] = Wr2[tid]; }
    const int c = tid & 63;
    for (int i = tid >> 6; i < NPG; i += 2)
        hn[i][c] = hin[(size_t)(base + i) * HID + c] * bn[c] + bn[64 + c];
    __syncthreads();

    if (tid < NPG) {
        float sL = 0.0f, sR = 0.0f;
        for (int cc = 0; cc < HID; ++cc) {
            sL += hn[tid][cc] * wl[cc];
            sR += hn[tid][cc] * wr[cc];
        }
        yl[tid] = sL + bl2[0];
        yr[tid] = sR + br2[0];
    }
    __syncthreads();

    const float a2 = att2[0];
    for (int t = tid; t < NPG * 17; t += 128) {
        const int i = t / 17, j = t - i * 17;
        const int s = (j < DEG) ? (esrc[(size_t)(base + i) * DEG + j] - base) : i;
        sl[i][j] = (unsigned short)s;
        float v = yl[s] + yr[i];
        v = (v > 0.0f) ? v : 0.2f * v;
        al[i][j] = a2 * v;
    }
    __syncthreads();

    if (tid < NPG) {
        float mx = -3.4e38f;
        for (int j = 0; j < 17; ++j) mx = fmaxf(mx, al[tid][j]);
        float den = 0.0f;
        for (int j = 0; j < 17; ++j) {
            const float e = expf(al[tid][j] - mx);
            al[tid][j] = e;                       // row owned by this thread
            den += e;
        }
        const float inv = 1.0f / (den + SM_EPS);
        float acc = 0.0f;
        for (int j = 0; j < 17; ++j)
            acc += al[tid][j] * inv * yl[sl[tid][j]];
        acc += bias2[0];
        // exact GELU: 0.5*x*(1+erf(x/sqrt(2)))
        out[(size_t)g * NPG + tid] =
            0.5f * acc * (1.0f + erff(acc * 0.70710678118654752f));
    }
}

// =====================================================================
extern "C" void kernel_launch(void* const* d_in, const int* in_sizes, int n_in,
                              void* d_out, int out_size, void* d_ws, size_t ws_size,
                              hipStream_t stream) {
    const float* x     = (const float*)d_in[0];
    const int*   eidx  = (const int*)  d_in[1];   // [2,E]; row 0 = src
    const float* Wl1   = (const float*)d_in[2];
    const float* bl1   = (const float*)d_in[3];
    const float* Wr1   = (const float*)d_in[4];
    const float* br1   = (const float*)d_in[5];
    const float* att1  = (const float*)d_in[6];
    const float* bias1 = (const float*)d_in[7];
    const float* gamma = (const float*)d_in[8];
    const float* beta  = (const float*)d_in[9];
    const float* Wl2   = (const float*)d_in[10];
    const float* bl2   = (const float*)d_in[11];
    const float* Wr2   = (const float*)d_in[12];
    const float* br2   = (const float*)d_in[13];
    const float* att2  = (const float*)d_in[14];
    const float* bias2 = (const float*)d_in[15];

    // workspace layout (floats)
    float* ws   = (float*)d_ws;
    float* h    = ws;                                   // N * 64
    float* psum = h + (size_t)NN * HID;                 // NG * 64
    float* psq  = psum + (size_t)NG * HID;              // NG * 64
    float* bn   = psq + (size_t)NG * HID;               // 128

    gat1_kernel<<<NG, 128, 0, stream>>>(x, eidx, Wl1, bl1, Wr1, br1,
                                        att1, bias1, h, psum, psq);
    bnstat_kernel<<<1, 128, 0, stream>>>(psum, psq, gamma, beta, bn);
    gat2_kernel<<<NG, 128, 0, stream>>>(h, eidx, bn, Wl2, bl2, Wr2, br2,
                                        att2, bias2, (float*)d_out);
}